// ScaledDotProductAttention_52398601011632
// MI455X (gfx1250) — compile-verified
//
#include <hip/hip_runtime.h>
#include <hip/hip_bf16.h>
#include <math.h>

typedef __attribute__((ext_vector_type(2))) float v2f;
typedef __attribute__((ext_vector_type(8))) float v8f;

#define INV_SCALE 0.08838834764831845f  /* 1/sqrt(128) */

constexpr int BB = 4, HH = 16, SS = 2048, DD = 128;
constexpr int QT  = 16;                 // query rows per workgroup
constexpr int SST = SS + 4;             // padded LDS row stride (floats): 2052 % 64 == 4
constexpr int RED_OFF = QT * SST;       // reduction scratch offset (floats)

__global__ __launch_bounds__(128) void fused_attn_kernel(
    const float* __restrict__ qg, const float* __restrict__ kg,
    const float* __restrict__ vg, const float* __restrict__ biasg,
    const float* __restrict__ loopg, float* __restrict__ outg,
    float* __restrict__ attng)
{
  extern __shared__ float smem[];

  const int tid  = threadIdx.x;
  const int lane = tid & 31;
  const int wave = tid >> 5;
  const int lh   = lane & 15;   // low-half lane index (M or N)
  const int hh   = lane >> 4;   // which half of the wave

  const int wg = blockIdx.x;
  const int qt = wg & (SS / QT - 1);      // 128 q-tiles
  const int h  = (wg >> 7) & (HH - 1);
  const int b  = wg >> 11;
  const int q_base = qt * QT;

  const size_t head_row = (size_t)(b * HH + h) * SS;
  const float* qp    = qg + (head_row + q_base) * DD;
  const float* kp    = kg + head_row * DD;
  const float* vp    = vg + head_row * DD;
  const float* biasp = biasg + ((size_t)h * SS + q_base) * SS;
  const float* loopp = loopg + (size_t)q_base * SS;
  float* outp  = outg  + (head_row + q_base) * DD;
  float* attnp = attng + (head_row + q_base) * SS;

  // ---------- Phase 1: S = (Q/scale) @ K^T ; S = (S + bias) * loop -> LDS ----------
  // A operand (16x4 f32): lane(0-15)=row M, vgpr0/1 = K{0,1}; lanes 16-31 = K{2,3}.
  v2f a[32];
  {
    const float* qrow = qp + lh * DD + hh * 2;
#pragma unroll
    for (int i = 0; i < 32; ++i) {
      v2f t = *(const v2f*)(qrow + i * 4);
      a[i] = t * INV_SCALE;
    }
  }

  for (int kt = wave * 32; kt < wave * 32 + 32; ++kt) {
    const int k_base = kt * QT;
    const float* krow = kp + (size_t)(k_base + lh) * DD + hh * 2;
    if (k_base + QT < SS)  // speculative prefetch of next K tile
      __builtin_prefetch(kp + (size_t)(k_base + QT + lh) * DD, 0, 1);

    v8f c = {};
#pragma unroll
    for (int i = 0; i < 32; ++i) {
      // B operand (4x16 f32): lane(0-15)=col N, vgpr0/1 = K rows {0,1}; hi half {2,3}.
      v2f bmat = *(const v2f*)(krow + i * 4);
      c = __builtin_amdgcn_wmma_f32_16x16x4_f32(false, a[i], false, bmat,
                                                (short)0, c, false, false);
    }
    // C layout: lane = hh*16 + N ; vgpr j -> M = j + hh*8
#pragma unroll
    for (int j = 0; j < 8; ++j) {
      const int m = j + hh * 8;
      const float bv = biasp[(size_t)m * SS + k_base + lh];
      const float lv = loopp[(size_t)m * SS + k_base + lh];
      smem[m * SST + k_base + lh] = (c[j] + bv) * lv;
    }
  }
  __syncthreads();

  // ---------- Phase 2: row softmax over keys; stream normalized P to global attn ----------
  {
    const int srow = tid >> 3;      // 16 rows, 8 threads each
    const int l8   = tid & 7;
    float* rowp = smem + srow * SST;
    float* red  = smem + RED_OFF;

    float mx = -3.402823466e38f;
#pragma unroll 4
    for (int j = 0; j < 64; ++j) {
      const int col = (l8 + 8 * j) * 4;
      float4 vv = *(const float4*)(rowp + col);
      mx = fmaxf(mx, fmaxf(fmaxf(vv.x, vv.y), fmaxf(vv.z, vv.w)));
    }
    red[srow * 8 + l8] = mx;
    __syncthreads();
    float rmx = red[srow * 8];
#pragma unroll
    for (int i = 1; i < 8; ++i) rmx = fmaxf(rmx, red[srow * 8 + i]);
    __syncthreads();  // protect red[] before reuse for sums

    float sum = 0.f;
#pragma unroll 4
    for (int j = 0; j < 64; ++j) {
      const int col = (l8 + 8 * j) * 4;
      float4 vv = *(const float4*)(rowp + col);
      vv.x = __expf(vv.x - rmx);
      vv.y = __expf(vv.y - rmx);
      vv.z = __expf(vv.z - rmx);
      vv.w = __expf(vv.w - rmx);
      *(float4*)(rowp + col) = vv;
      sum += vv.x + vv.y + vv.z + vv.w;
    }
    red[srow * 8 + l8] = sum;
    __syncthreads();
    float rs = 0.f;
#pragma unroll
    for (int i = 0; i < 8; ++i) rs += red[srow * 8 + i];
    const float inv = 1.0f / rs;

    float* arow = attnp + (size_t)srow * SS;
#pragma unroll 4
    for (int j = 0; j < 64; ++j) {
      const int col = (l8 + 8 * j) * 4;
      float4 vv = *(const float4*)(rowp + col);
      vv.x *= inv; vv.y *= inv; vv.z *= inv; vv.w *= inv;
      *(float4*)(rowp + col) = vv;      // keep normalized P in LDS for PV pass
      *(float4*)(arow + col) = vv;      // coalesced 128B global attn store
    }
  }
  __syncthreads();

  // ---------- Phase 3: O = P @ V ----------
#pragma unroll
  for (int t = 0; t < 2; ++t) {
    const int ct = wave * 2 + t;              // 16-wide output column tile
    const float* vcol = vp + ct * 16 + lh;    // V[key][d], lane = d column
    const float* prow = smem + lh * SST + hh * 2;  // padded -> conflict-free b64 reads
    v8f c = {};
#pragma unroll 8
    for (int kk = 0; kk < SS / 4; ++kk) {
      const int k0 = kk * 4;
      v2f aa = *(const v2f*)(prow + k0);      // A: P rows, K = k0 + hh*2 + {0,1}
      v2f bb;                                 // B: V rows k0+hh*2+{0,1}, col = lh
      bb.x = vcol[(size_t)(k0 + hh * 2) * DD];
      bb.y = vcol[(size_t)(k0 + hh * 2 + 1) * DD];
      c = __builtin_amdgcn_wmma_f32_16x16x4_f32(false, aa, false, bb,
                                               (short)0, c, false, false);
    }
#pragma unroll
    for (int j = 0; j < 8; ++j) {
      const int m = j + hh * 8;
      outp[(size_t)m * DD + ct * 16 + lh] = c[j];
    }
  }
}

extern "C" void kernel_launch(void* const* d_in, const int* in_sizes, int n_in,
                              void* d_out, int out_size, void* d_ws, size_t ws_size,
                              hipStream_t stream) {
  const float* q    = (const float*)d_in[0];
  const float* k    = (const float*)d_in[1];
  const float* v    = (const float*)d_in[2];
  const float* bias = (const float*)d_in[3];
  const float* loop = (const float*)d_in[4];
  float* out  = (float*)d_out;
  float* attn = (float*)d_out + (size_t)BB * HH * SS * DD;  // tuple order: (output, attn)

  const size_t lds_bytes = (size_t)(RED_OFF + QT * 8) * sizeof(float);  // ~131.9 KB
  dim3 grid(BB * HH * (SS / QT));  // 8192 workgroups
  fused_attn_kernel<<<grid, 128, lds_bytes, stream>>>(q, k, v, bias, loop, out, attn);

  (void)in_sizes; (void)n_in; (void)out_size; (void)d_ws; (void)ws_size;
}